// ConvexExpansionAttention_44220983279911
// MI455X (gfx1250) — compile-verified
//
#include <hip/hip_runtime.h>
#include <hip/hip_bf16.h>
#include <math.h>

// ---------------------------------------------------------------------------
// ConvexExpansionAttention for MI455X (gfx1250, wave32)
//
// Sizes (from reference): B=2, T=1024 -> NT=2048 tokens, d=128, D3=384, D9=1152
// Algebra: (x @ softplus(W).T + b) @ S  ==  x @ (softplus(W).T @ S) + b @ S
// so we fuse each head's two GEMMs into one 128x384 matrix (precomputed per
// launch; deterministic).  GEMMs use V_WMMA_F32_16X16X4_F32 (f32-exact).
// ---------------------------------------------------------------------------

#define NT     2048           // B*T tokens
#define DD     384            // D
#define DH     128            // d
#define D3     384
#define D9     1152
#define GAMMA  5.0f
#define LOG2E  1.4426950408889634f

typedef float v2f __attribute__((ext_vector_type(2)));
typedef float v8f __attribute__((ext_vector_type(8)));

static __device__ __forceinline__ v8f wmma4(v2f a, v2f b, v8f c) {
  // D = A(16x4 f32) * B(4x16 f32) + C(16x16 f32)
  return __builtin_amdgcn_wmma_f32_16x16x4_f32(
      /*neg_a=*/false, a, /*neg_b=*/false, b,
      /*c_mod=*/(short)0, c, /*reuse_a=*/false, /*reuse_b=*/false);
}

static __device__ __forceinline__ float softplusf(float w) {
  return (w > 20.0f) ? w : log1pf(__expf(w));
}

// --------------------------- weight preprocessing ---------------------------

// out[i] = softplus(in[i])
__global__ void cea_softplus(const float* __restrict__ in,
                             float* __restrict__ out, int n) {
  int i = blockIdx.x * blockDim.x + threadIdx.x;
  if (i < n) out[i] = softplusf(in[i]);
}

// Mt[h][n][k] = sum_j softplus(W_h)[j][k] * S_h[j][n]   (n-major for WMMA B)
__global__ void cea_fuse(const float* __restrict__ spw,   // 3 x [384][128]
                         const float* __restrict__ Sq,
                         const float* __restrict__ Sk,
                         const float* __restrict__ Sv,
                         float* __restrict__ Mt) {        // 3 x [384][128]
  int h   = blockIdx.y;
  int idx = blockIdx.x * 256 + threadIdx.x;   // idx = n*128 + k
  int n = idx >> 7, k = idx & 127;
  const float* S = (h == 0) ? Sq : (h == 1) ? Sk : Sv;
  const float* W = spw + h * (D3 * DH);
  float acc = 0.0f;
  for (int j = 0; j < D3; ++j) acc += W[j * DH + k] * S[j * D3 + n];
  Mt[h * (D3 * DH) + idx] = acc;
}

// c[h][n] = sum_j b_h[j] * S_h[j][n]
__global__ void cea_bias(const float* __restrict__ bq, const float* __restrict__ bk,
                         const float* __restrict__ bv,
                         const float* __restrict__ Sq, const float* __restrict__ Sk,
                         const float* __restrict__ Sv,
                         float* __restrict__ c) {
  int h = blockIdx.x, n = threadIdx.x;
  const float* b = (h == 0) ? bq : (h == 1) ? bk : bv;
  const float* S = (h == 0) ? Sq : (h == 1) ? Sk : Sv;
  float acc = 0.0f;
  for (int j = 0; j < D3; ++j) acc += b[j] * S[j * D3 + n];
  c[h * D3 + n] = acc;
}

// ------------------------------ QKV GEMM (WMMA) -----------------------------
// qkv[t][h*384+n] = sum_k x[t][h*128+k] * Mt[h][n][k] + c[h][n]
// block: 256 threads (8 waves); each block = 16 tokens x 384 cols of one head.
// Each wave owns 3 N-tiles (8*3*16 = 384 cols).  K = 128 -> 32 wmma steps.
__global__ void cea_qkv(const float* __restrict__ x,
                        const float* __restrict__ Mt,
                        const float* __restrict__ cb,
                        float* __restrict__ qkv) {
  __shared__ float xT[16 * DH];               // 8 KB: A tile, shared by 8 waves
  const int h  = blockIdx.y;
  const int t0 = blockIdx.x * 16;
  const int tid = threadIdx.x;

  for (int idx = tid; idx < 16 * DH; idx += 256) {
    int r = idx >> 7, cc = idx & 127;
    xT[idx] = x[(t0 + r) * DD + h * DH + cc];
  }
  __syncthreads();

  const int lane = tid & 31, wid = tid >> 5;
  const int m  = lane & 15;                   // A row / B col within tile
  const int kh = (lane >> 4) * 2;             // K sub-offset {0,2}
  const float* __restrict__ Mh = Mt + h * (D3 * DH);
  const int n0 = wid * 48;                    // 3 tiles of 16 per wave

  v8f acc0 = {}, acc1 = {}, acc2 = {};
  for (int k0 = 0; k0 < DH; k0 += 4) {
    v2f a  = *(const v2f*)&xT[m * DH + k0 + kh];
    v2f b0 = *(const v2f*)&Mh[(n0 + 0  + m) * DH + k0 + kh];
    v2f b1 = *(const v2f*)&Mh[(n0 + 16 + m) * DH + k0 + kh];
    v2f b2 = *(const v2f*)&Mh[(n0 + 32 + m) * DH + k0 + kh];
    acc0 = wmma4(a, b0, acc0);
    acc1 = wmma4(a, b1, acc1);
    acc2 = wmma4(a, b2, acc2);
  }

  const int rbase = (lane >> 4) * 8;          // C layout: rows r and r+8
  const int ncol  = lane & 15;
  for (int r = 0; r < 8; ++r) {
    int row = t0 + rbase + r;
    qkv[row * D9 + h * D3 + n0 + 0  + ncol] = acc0[r] + cb[h * D3 + n0 + 0  + ncol];
    qkv[row * D9 + h * D3 + n0 + 16 + ncol] = acc1[r] + cb[h * D3 + n0 + 16 + ncol];
    qkv[row * D9 + h * D3 + n0 + 32 + ncol] = acc2[r] + cb[h * D3 + n0 + 32 + ncol];
  }
}

// --------------------------- per-token attention ----------------------------
// attended[t][i] = sum_j sig(g*Q_i*K_j)*V_j / (sum_j sig(g*Q_i*K_j) + eps)
// One block per token, 384 threads (one per output row i).  Q/K/V in LDS;
// K pre-scaled by -gamma*log2e so sigmoid = rcp(1 + exp2(Qi * Ksc[j])).
__global__ void cea_attn(const float* __restrict__ qkv,
                         float* __restrict__ att) {
  __shared__ float Qs[D3], Ksc[D3], Vs[D3];
  const int t = blockIdx.x, i = threadIdx.x;
  const float* __restrict__ base = qkv + t * D9;
  Qs[i]  = base[i];
  Ksc[i] = base[D3 + i] * (-GAMMA * LOG2E);
  Vs[i]  = base[2 * D3 + i];
  __syncthreads();

  const float Qi = Qs[i];
  float acc = 0.0f, ssum = 0.0f;
#pragma unroll 4
  for (int j = 0; j < D3; ++j) {
    float s = __builtin_amdgcn_rcpf(1.0f + __builtin_amdgcn_exp2f(Qi * Ksc[j]));
    acc  += s * Vs[j];
    ssum += s;
  }
  att[t * D3 + i] = acc / (ssum + 1e-8f);
}

// --------------------------- projection GEMM (WMMA) -------------------------
// h[t][n] = sum_i att[t][i] * softplus(Wp)[n][i] + bp[n]  -> written raw to out
// block: 256 threads, 16 tokens x 1152 cols; 8 waves x 9 N-tiles; K=384.
__global__ void cea_proj(const float* __restrict__ att,
                         const float* __restrict__ spP,   // [1152][384], n-major
                         const float* __restrict__ bp,
                         float* __restrict__ hout) {
  __shared__ float aT[16 * D3];               // 24 KB A tile
  const int t0 = blockIdx.x * 16;
  const int tid = threadIdx.x;
  for (int idx = tid; idx < 16 * D3; idx += 256)
    aT[idx] = att[t0 * D3 + idx];             // 16 consecutive token rows
  __syncthreads();

  const int lane = tid & 31, wid = tid >> 5;
  const int m  = lane & 15;
  const int kh = (lane >> 4) * 2;
  const int nbase = wid * 144;                // 9 tiles * 16 cols per wave

  v8f acc[9];
#pragma unroll
  for (int q = 0; q < 9; ++q) acc[q] = (v8f){};

  for (int k0 = 0; k0 < D3; k0 += 4) {
    v2f a = *(const v2f*)&aT[m * D3 + k0 + kh];
#pragma unroll
    for (int q = 0; q < 9; ++q) {
      v2f b = *(const v2f*)&spP[(nbase + q * 16 + m) * D3 + k0 + kh];
      acc[q] = wmma4(a, b, acc[q]);
    }
  }

  const int rbase = (lane >> 4) * 8;
  const int ncol  = lane & 15;
#pragma unroll
  for (int q = 0; q < 9; ++q) {
    int n = nbase + q * 16 + ncol;
    float bias = bp[n];
    for (int r = 0; r < 8; ++r)
      hout[(t0 + rbase + r) * D9 + n] = acc[q][r] + bias;
  }
}

// ------------------------------- layernorm ----------------------------------
// In-place over d_out: out = (h - mu) * rsqrt(var + eps) * gn + bn, per token.
__global__ void cea_ln(float* __restrict__ out,
                       const float* __restrict__ gn,
                       const float* __restrict__ bn) {
  __shared__ float row[D9];
  __shared__ float red[256];
  const int t = blockIdx.x, tid = threadIdx.x;
  float* __restrict__ h = out + t * D9;

  float ps = 0.0f;
  for (int n = tid; n < D9; n += 256) { float v = h[n]; row[n] = v; ps += v; }
  red[tid] = ps;
  __syncthreads();
  for (int s = 128; s > 0; s >>= 1) {
    if (tid < s) red[tid] += red[tid + s];
    __syncthreads();
  }
  const float mu = red[0] * (1.0f / (float)D9);
  __syncthreads();

  float pv = 0.0f;
  for (int n = tid; n < D9; n += 256) { float dv = row[n] - mu; pv += dv * dv; }
  red[tid] = pv;
  __syncthreads();
  for (int s = 128; s > 0; s >>= 1) {
    if (tid < s) red[tid] += red[tid + s];
    __syncthreads();
  }
  const float var = red[0] * (1.0f / (float)D9);
  const float rs  = __builtin_amdgcn_rsqf(var + 1e-5f);

  for (int n = tid; n < D9; n += 256)
    h[n] = (row[n] - mu) * rs * gn[n] + bn[n];
}

// --------------------------------- launch -----------------------------------

extern "C" void kernel_launch(void* const* d_in, const int* in_sizes, int n_in,
                              void* d_out, int out_size, void* d_ws, size_t ws_size,
                              hipStream_t stream) {
  const float* x  = (const float*)d_in[0];
  const float* Wq = (const float*)d_in[1];
  const float* bq = (const float*)d_in[2];
  const float* Sq = (const float*)d_in[3];
  const float* Wk = (const float*)d_in[4];
  const float* bk = (const float*)d_in[5];
  const float* Sk = (const float*)d_in[6];
  const float* Wv = (const float*)d_in[7];
  const float* bv = (const float*)d_in[8];
  const float* Sv = (const float*)d_in[9];
  const float* Wp = (const float*)d_in[10];
  const float* bp = (const float*)d_in[11];
  const float* gn = (const float*)d_in[12];
  const float* bn = (const float*)d_in[13];
  float* out = (float*)d_out;

  // workspace layout (floats); all offsets even -> 8B-aligned float2 loads
  float* ws = (float*)d_ws;
  float* spw  = ws;                               // 3 * 384*128  = 147456
  float* Mt   = spw + 3 * D3 * DH;                // 3 * 384*128  = 147456
  float* cb   = Mt  + 3 * D3 * DH;                // 3 * 384      = 1152
  float* spP  = cb  + 3 * D3;                     // 1152*384     = 442368
  float* qkv  = spP + D9 * D3;                    // 2048*1152    = 2359296
  float* attb = qkv + NT * D9;                    // 2048*384     = 786432

  // 1) softplus of all weight matrices
  cea_softplus<<<dim3((D3 * DH + 255) / 256), dim3(256), 0, stream>>>(Wq, spw,              D3 * DH);
  cea_softplus<<<dim3((D3 * DH + 255) / 256), dim3(256), 0, stream>>>(Wk, spw + D3 * DH,    D3 * DH);
  cea_softplus<<<dim3((D3 * DH + 255) / 256), dim3(256), 0, stream>>>(Wv, spw + 2 * D3 * DH, D3 * DH);
  cea_softplus<<<dim3((D9 * D3 + 255) / 256), dim3(256), 0, stream>>>(Wp, spP,              D9 * D3);

  // 2) fuse softplus(W)^T @ S (n-major) and rotate biases
  cea_fuse<<<dim3((D3 * DH) / 256, 3), dim3(256), 0, stream>>>(spw, Sq, Sk, Sv, Mt);
  cea_bias<<<dim3(3), dim3(D3), 0, stream>>>(bq, bk, bv, Sq, Sk, Sv, cb);

  // 3) QKV GEMM via V_WMMA_F32_16X16X4_F32
  cea_qkv<<<dim3(NT / 16, 3), dim3(256), 0, stream>>>(x, Mt, cb, qkv);

  // 4) per-token sigmoid attention (VALU + v_exp/v_rcp)
  cea_attn<<<dim3(NT), dim3(D3), 0, stream>>>(qkv, attb);

  // 5) output projection via WMMA -> raw h into d_out
  cea_proj<<<dim3(NT / 16), dim3(256), 0, stream>>>(attb, spP, bp, out);

  // 6) in-place layernorm
  cea_ln<<<dim3(NT), dim3(256), 0, stream>>>(out, gn, bn);

  (void)in_sizes; (void)n_in; (void)out_size; (void)ws_size;
}